// TopicConvCausalLayer0_56581899157987
// MI455X (gfx1250) — compile-verified
//
#include <hip/hip_runtime.h>

typedef __attribute__((ext_vector_type(2))) float v2f;
typedef __attribute__((ext_vector_type(8))) float v8f;

#define DDIM 256
#define NWORD 20000
#define NTOPIC 1000

// ---------------------------------------------------------------- utilities
__global__ __launch_bounds__(256) void zero_f(float* __restrict__ p, int n) {
  int i = blockIdx.x * 256 + threadIdx.x;
  if (i < n) p[i] = 0.0f;
}

__device__ __forceinline__ float sgnf(float x) {
  return (x > 0.0f) ? 1.0f : ((x < 0.0f) ? -1.0f : 0.0f);
}

// ------------------------------------------------- word GEMM: Wh_ww & Wh_wt
// One wave computes a 80(M)x16(N) strip of BOTH outputs: 5 M-tiles share the
// B operands, so each K-step is 5 A-loads + 2 B-pairs -> 10 WMMAs.
// 1250 M-tiles = 250 groups * 16 N-tiles = 4000 wave-tiles -> 500 blocks.
__global__ __launch_bounds__(256) void gemm_word(
    const float* __restrict__ X,      // [20000,256]
    const float* __restrict__ Www, const float* __restrict__ bww,
    const float* __restrict__ Wwt, const float* __restrict__ bwt,
    float* __restrict__ Yww, float* __restrict__ Ywt)
{
  const int wave = threadIdx.x >> 5;
  const int lane = threadIdx.x & 31;
  const int tile = blockIdx.x * 8 + wave;     // 0..3999
  const int g  = tile >> 4;                   // M-group (80 rows) 0..249
  const int tn = tile & 15;                   // N-tile 0..15
  const int half = lane >> 4;                 // 0: K={0,1}, 1: K={2,3}
  const int lm = lane & 15;
  const int col = tn * 16 + lm;

  v8f accw[5] = {};
  v8f acct[5] = {};
  const float* __restrict__ ap = X + (size_t)(g * 80 + lm) * DDIM;

  for (int kb = 0; kb < DDIM / 4; ++kb) {
    const int k0 = kb * 4 + half * 2;
    v2f a[5];
#pragma unroll
    for (int i = 0; i < 5; ++i)
      a[i] = *(const v2f*)(ap + (size_t)i * 16 * DDIM + k0);
    v2f b0, b1;
    b0[0] = Www[(size_t)k0 * DDIM + col];
    b0[1] = Www[(size_t)(k0 + 1) * DDIM + col];
    b1[0] = Wwt[(size_t)k0 * DDIM + col];
    b1[1] = Wwt[(size_t)(k0 + 1) * DDIM + col];
#pragma unroll
    for (int i = 0; i < 5; ++i) {
      accw[i] = __builtin_amdgcn_wmma_f32_16x16x4_f32(false, a[i], false, b0, (short)0, accw[i], false, false);
      acct[i] = __builtin_amdgcn_wmma_f32_16x16x4_f32(false, a[i], false, b1, (short)0, acct[i], false, false);
    }
  }

  const float biasw = bww[col];
  const float biast = bwt[col];
#pragma unroll
  for (int i = 0; i < 5; ++i) {
#pragma unroll
    for (int r = 0; r < 8; ++r) {
      const int orow = g * 80 + i * 16 + half * 8 + r;  // C/D: VGPR r -> M=r (+8 upper half)
      Yww[(size_t)orow * DDIM + col] = accw[i][r] + biasw;
      Ywt[(size_t)orow * DDIM + col] = acct[i][r] + biast;
    }
  }
}

// --------------------------------------------- topic GEMM: Wh_tt (+ causal)
// Wh_tt = topic@W_tt + b_tt + (1/3) * sum_t sign(effect[n,t]) * (topic@Wcau_t)
// 63 M-tiles -> 21 groups of 3; 4 B chains shared across the 3 M-tiles:
// each K-step is 3 A-loads + 4 B-pairs -> 12 WMMAs.
// 21 * 16 = 336 wave-tiles -> 42 blocks. Last group row-clamped/masked.
__global__ __launch_bounds__(256) void gemm_topic(
    const float* __restrict__ X,       // [1000,256]
    const float* __restrict__ Wtt, const float* __restrict__ btt,
    const float* __restrict__ Wcau,    // [3,256,256]
    const float* __restrict__ effect,  // [1000,3]
    float* __restrict__ Ytt)
{
  const int wave = threadIdx.x >> 5;
  const int lane = threadIdx.x & 31;
  const int tile = blockIdx.x * 8 + wave;     // 0..335
  const int g  = tile >> 4;                   // M-group (48 rows) 0..20
  const int tn = tile & 15;
  const int half = lane >> 4;
  const int lm = lane & 15;
  const int col = tn * 16 + lm;

  v8f acc[3][4] = {};
  const float* __restrict__ Wc0 = Wcau;
  const float* __restrict__ Wc1 = Wcau + (size_t)DDIM * DDIM;
  const float* __restrict__ Wc2 = Wcau + (size_t)2 * DDIM * DDIM;

  int arow[3];
#pragma unroll
  for (int i = 0; i < 3; ++i) {
    int r = (g * 3 + i) * 16 + lm;
    arow[i] = (r > NTOPIC - 1) ? (NTOPIC - 1) : r;   // clamp loads; EXEC stays full
  }

  for (int kb = 0; kb < DDIM / 4; ++kb) {
    const int k0 = kb * 4 + half * 2;
    v2f a[3];
#pragma unroll
    for (int i = 0; i < 3; ++i)
      a[i] = *(const v2f*)(X + (size_t)arow[i] * DDIM + k0);
    const size_t o0 = (size_t)k0 * DDIM + col;
    const size_t o1 = (size_t)(k0 + 1) * DDIM + col;
    v2f bt, b0, b1, b2;
    bt[0] = Wtt[o0]; bt[1] = Wtt[o1];
    b0[0] = Wc0[o0]; b0[1] = Wc0[o1];
    b1[0] = Wc1[o0]; b1[1] = Wc1[o1];
    b2[0] = Wc2[o0]; b2[1] = Wc2[o1];
#pragma unroll
    for (int i = 0; i < 3; ++i) {
      acc[i][0] = __builtin_amdgcn_wmma_f32_16x16x4_f32(false, a[i], false, bt, (short)0, acc[i][0], false, false);
      acc[i][1] = __builtin_amdgcn_wmma_f32_16x16x4_f32(false, a[i], false, b0, (short)0, acc[i][1], false, false);
      acc[i][2] = __builtin_amdgcn_wmma_f32_16x16x4_f32(false, a[i], false, b1, (short)0, acc[i][2], false, false);
      acc[i][3] = __builtin_amdgcn_wmma_f32_16x16x4_f32(false, a[i], false, b2, (short)0, acc[i][3], false, false);
    }
  }

  const float bias = btt[col];
#pragma unroll
  for (int i = 0; i < 3; ++i) {
#pragma unroll
    for (int r = 0; r < 8; ++r) {
      const int orow = (g * 3 + i) * 16 + half * 8 + r;
      if (orow < NTOPIC) {
        const float* ef = effect + (size_t)orow * 3;
        const float s0 = sgnf(ef[0]);
        const float s1 = sgnf(ef[1]);
        const float s2 = sgnf(ef[2]);
        Ytt[(size_t)orow * DDIM + col] =
            acc[i][0][r] + bias +
            (s0 * acc[i][1][r] + s1 * acc[i][2][r] + s2 * acc[i][3][r]) * (1.0f / 3.0f);
      }
    }
  }
}

// ----------------------------------------- edge scatter: sum += Wh[src]*w
// One thread per (edge, 4-feature chunk): 64 threads/edge, b128 gather,
// 4x global_atomic_add_f32 scatter. q==0 thread bumps the edge count.
__global__ __launch_bounds__(256) void scatter_edges(
    const float* __restrict__ Wh, const float* __restrict__ ew,
    const int* __restrict__ src, const int* __restrict__ dst,
    float* __restrict__ sum, float* __restrict__ cnt, int E)
{
  const long long t = (long long)blockIdx.x * 256 + threadIdx.x;
  const int e = (int)(t >> 6);
  const int q = (int)(t & 63);
  if (e >= E) return;
  const int s = src[e];
  const int d = dst[e];
  const float w = ew[e];
  const float4 v = ((const float4*)(Wh + (size_t)s * DDIM))[q];
  float* o = sum + (size_t)d * DDIM + q * 4;
  atomicAdd(o + 0, v.x * w);
  atomicAdd(o + 1, v.y * w);
  atomicAdd(o + 2, v.z * w);
  atomicAdd(o + 3, v.w * w);
  if (q == 0) atomicAdd(cnt + d, 1.0f);
}

// ------------------------------------------------------------- finalizers
__global__ __launch_bounds__(256) void finalize_word(float* __restrict__ out,
                                                     const float* __restrict__ cnt) {
  const int i = blockIdx.x * 256 + threadIdx.x;        // 5,120,000 elems
  out[i] = out[i] / fmaxf(cnt[i >> 8], 1.0f);
}

__global__ __launch_bounds__(256) void finalize_topic(
    float* __restrict__ out,
    const float* __restrict__ swt, const float* __restrict__ cwt,
    const float* __restrict__ stt, const float* __restrict__ ctt) {
  const int i = blockIdx.x * 256 + threadIdx.x;        // 256,000 elems
  const int node = i >> 8;
  out[i] = swt[i] / fmaxf(cwt[node], 1.0f) + stt[i] / fmaxf(ctt[node], 1.0f);
}

// ------------------------------------------------------------------ launch
extern "C" void kernel_launch(void* const* d_in, const int* in_sizes, int n_in,
                              void* d_out, int out_size, void* d_ws, size_t ws_size,
                              hipStream_t stream) {
  const float* word_feat  = (const float*)d_in[0];
  const float* topic_feat = (const float*)d_in[1];
  const float* effect     = (const float*)d_in[2];
  const float* W_ww  = (const float*)d_in[3];
  const float* b_ww  = (const float*)d_in[4];
  const float* W_wt  = (const float*)d_in[5];
  const float* b_wt  = (const float*)d_in[6];
  const float* W_tt  = (const float*)d_in[7];
  const float* b_tt  = (const float*)d_in[8];
  const float* W_cau = (const float*)d_in[9];
  const float* ww_w  = (const float*)d_in[10];
  const float* wt_w  = (const float*)d_in[11];
  const float* tt_w  = (const float*)d_in[12];
  const int* ww_src = (const int*)d_in[13];
  const int* ww_dst = (const int*)d_in[14];
  const int* wt_src = (const int*)d_in[15];
  const int* wt_dst = (const int*)d_in[16];
  const int* tt_src = (const int*)d_in[17];
  const int* tt_dst = (const int*)d_in[18];

  const int E_ww = in_sizes[13];
  const int E_wt = in_sizes[15];
  const int E_tt = in_sizes[17];

  float* out      = (float*)d_out;
  float* outWord  = out;                        // [20000*256] used as sum then divided
  float* outTopic = out + (size_t)NWORD * DDIM; // [1000*256]

  // workspace layout (floats)
  float* ws = (float*)d_ws;
  float* Wh_ww  = ws;                   // 5,120,000
  float* Wh_wt  = ws + 5120000;         // 5,120,000
  float* Wh_tt  = ws + 10240000;        //   256,000
  float* sum_wt = ws + 10496000;        //   256,000
  float* sum_tt = ws + 10752000;        //   256,000
  float* cnt_ww = ws + 11008000;        //    20,000
  float* cnt_wt = ws + 11028000;        //     1,000
  float* cnt_tt = ws + 11029000;        //     1,000

  // zero accumulation targets (harness does not re-zero between replays)
  zero_f<<<(NWORD * DDIM + 255) / 256, 256, 0, stream>>>(outWord, NWORD * DDIM);
  zero_f<<<(NTOPIC * DDIM + 255) / 256, 256, 0, stream>>>(sum_wt, NTOPIC * DDIM);
  zero_f<<<(NTOPIC * DDIM + 255) / 256, 256, 0, stream>>>(sum_tt, NTOPIC * DDIM);
  zero_f<<<(NWORD + 255) / 256, 256, 0, stream>>>(cnt_ww, NWORD);
  zero_f<<<(NTOPIC + 255) / 256, 256, 0, stream>>>(cnt_wt, NTOPIC);
  zero_f<<<(NTOPIC + 255) / 256, 256, 0, stream>>>(cnt_tt, NTOPIC);

  // dense projections on the WMMA pipe
  gemm_word<<<500, 256, 0, stream>>>(word_feat, W_ww, b_ww, W_wt, b_wt, Wh_ww, Wh_wt);
  gemm_topic<<<42, 256, 0, stream>>>(topic_feat, W_tt, b_tt, W_cau, effect, Wh_tt);

  // edge aggregation (u_mul_e + per-etype mean pieces)
  {
    long long t = (long long)E_ww * 64;
    scatter_edges<<<(unsigned)((t + 255) / 256), 256, 0, stream>>>(Wh_ww, ww_w, ww_src, ww_dst,
                                                                   outWord, cnt_ww, E_ww);
  }
  {
    long long t = (long long)E_wt * 64;
    scatter_edges<<<(unsigned)((t + 255) / 256), 256, 0, stream>>>(Wh_wt, wt_w, wt_src, wt_dst,
                                                                   sum_wt, cnt_wt, E_wt);
  }
  {
    long long t = (long long)E_tt * 64;
    scatter_edges<<<(unsigned)((t + 255) / 256), 256, 0, stream>>>(Wh_tt, tt_w, tt_src, tt_dst,
                                                                   sum_tt, cnt_tt, E_tt);
  }

  // means (+ cross-etype sum at topic nodes)
  finalize_word<<<NWORD * DDIM / 256, 256, 0, stream>>>(outWord, cnt_ww);
  finalize_topic<<<NTOPIC * DDIM / 256, 256, 0, stream>>>(outTopic, sum_wt, cnt_wt, sum_tt, cnt_tt);
}